// ConsistencyLoss_28913719837033
// MI455X (gfx1250) — compile-verified
//
#include <hip/hip_runtime.h>
#include <hip/hip_bf16.h>

typedef __attribute__((ext_vector_type(2))) float v2f;
typedef __attribute__((ext_vector_type(8))) float v8f;

// Fixed problem shape from the reference: [16, 1, 512, 1024] fp32.
constexpr int Hc = 512;
constexpr int Wc = 1024;

__device__ __forceinline__ float bilinear_zero(const float* __restrict__ im,
                                               float x, float y) {
    // grid_sample, zero padding, align_corners=False semantics (matches ref).
    float x0f = floorf(x);
    float y0f = floorf(y);
    float wx = x - x0f;
    float wy = y - y0f;
    int x0 = (int)x0f;
    int y0 = (int)y0f;
    int x1 = x0 + 1;
    int y1 = y0 + 1;

    auto g = [&](int yi, int xi) -> float {
        bool inb = (xi >= 0) & (xi < Wc) & (yi >= 0) & (yi < Hc);
        int xc = min(max(xi, 0), Wc - 1);
        int yc = min(max(yi, 0), Hc - 1);
        float v = im[yc * Wc + xc];
        return inb ? v : 0.0f;
    };

    float v00 = g(y0, x0);
    float v01 = g(y0, x1);
    float v10 = g(y1, x0);
    float v11 = g(y1, x1);
    return v00 * (1.0f - wx) * (1.0f - wy) + v01 * wx * (1.0f - wy) +
           v10 * (1.0f - wx) * wy + v11 * wx * wy;
}

// Pass 1: per-pixel warp + |diff| for both directions, block-level
// deterministic reduction to one partial per block.
__global__ void __launch_bounds__(256)
lr_consistency_partial(const float* __restrict__ ld,   // left_disp
                       const float* __restrict__ rd,   // right_disp
                       float* __restrict__ partial,
                       long long total) {
    const float invWm1 = 1.0f / (float)(Wc - 1);
    const float hscale = (float)Hc / (float)(Hc - 1);

    long long tid = (long long)blockIdx.x * blockDim.x + threadIdx.x;
    long long stride = (long long)gridDim.x * blockDim.x;

    float acc = 0.0f;
    for (long long p = tid; p < total; p += stride) {
        int j = (int)(p & (Wc - 1));          // x
        int i = (int)((p >> 10) & (Hc - 1));  // y  (Wc = 1024 = 2^10)
        int b = (int)(p >> 19);               // batch (Hc*Wc = 2^19)

        const float* ldp = ld + ((long long)b << 19);
        const float* rdp = rd + ((long long)b << 19);

        float dl = ldp[(i << 10) + j];
        float dr = rdp[(i << 10) + j];

        float xb = (float)j * invWm1;
        float y  = (float)i * hscale - 0.5f;

        // left_lr = warp(right_disp, left_disp, sign=-1)
        float xl = (xb - dl) * (float)Wc - 0.5f;
        float left_lr = bilinear_zero(rdp, xl, y);

        // right_lr = warp(left_disp, right_disp, sign=+1)
        float xr = (xb + dr) * (float)Wc - 0.5f;
        float right_lr = bilinear_zero(ldp, xr, y);

        acc += fabsf(dl - left_lr) + fabsf(dr - right_lr);
    }

    // Deterministic in-wave reduce (wave32), then LDS across the 8 waves.
    for (int s = 16; s > 0; s >>= 1) acc += __shfl_down(acc, s, 32);

    __shared__ float wsum[8];
    int lane = threadIdx.x & 31;
    int wave = threadIdx.x >> 5;
    if (lane == 0) wsum[wave] = acc;
    __syncthreads();
    if (threadIdx.x == 0) {
        float s = 0.0f;
        int nw = blockDim.x >> 5;
        for (int w = 0; w < nw; ++w) s += wsum[w];
        partial[blockIdx.x] = s;
    }
}

// Pass 2: single block reduces n partials deterministically; the last 64
// values are summed with one V_WMMA_F32_16X16X4_F32 (A = values, B = ones,
// C = 0).  D[m][n] = rowsum_m for every n, so summing D column 0 over all
// m (lane 0 holds M=0..7, lane 16 holds M=8..15) gives the sum of all 64
// A entries independent of the exact lane<->(m,k) mapping.
__global__ void __launch_bounds__(256)
lr_consistency_final(const float* __restrict__ partial, int n,
                     float* __restrict__ out, float invN) {
    __shared__ float sdata[256];

    float acc = 0.0f;
    for (int i = threadIdx.x; i < n; i += 256) acc += partial[i];
    sdata[threadIdx.x] = acc;
    __syncthreads();

    if (threadIdx.x < 32) {  // wave 0: EXEC all-ones for the WMMA
        int lane = threadIdx.x;
        float v0 = 0.0f, v1 = 0.0f;
        #pragma unroll
        for (int k = 0; k < 4; ++k) {
            v0 += sdata[lane * 8 + k];
            v1 += sdata[lane * 8 + 4 + k];
        }
        v2f a = {v0, v1};
        v2f bones = {1.0f, 1.0f};
        v8f c = {};
        // (neg_a, A, neg_b, B, c_mod, C, reuse_a, reuse_b)
        c = __builtin_amdgcn_wmma_f32_16x16x4_f32(
                false, a, false, bones, (short)0, c, false, false);
        float rs = c[0] + c[1] + c[2] + c[3] + c[4] + c[5] + c[6] + c[7];
        sdata[lane] = rs;
    }
    __syncthreads();

    if (threadIdx.x == 0) {
        out[0] = (sdata[0] + sdata[16]) * invN;
    }
}

extern "C" void kernel_launch(void* const* d_in, const int* in_sizes, int n_in,
                              void* d_out, int out_size, void* d_ws, size_t ws_size,
                              hipStream_t stream) {
    const float* left_disp  = (const float*)d_in[0];
    const float* right_disp = (const float*)d_in[1];
    float* out = (float*)d_out;

    long long total = (long long)in_sizes[0];  // B*1*H*W = 16*512*1024
    float invN = 1.0f / (float)total;

    int NB = 2048;
    size_t cap = ws_size / sizeof(float);
    if ((size_t)NB > cap) NB = (int)cap;
    if (NB < 1) NB = 1;

    float* partial = (float*)d_ws;

    lr_consistency_partial<<<NB, 256, 0, stream>>>(left_disp, right_disp,
                                                   partial, total);
    lr_consistency_final<<<1, 256, 0, stream>>>(partial, NB, out, invN);
}